// SimpleLTMBank_62594853372105
// MI455X (gfx1250) — compile-verified
//
#include <hip/hip_runtime.h>
#include <cfloat>
#include <climits>

typedef __attribute__((ext_vector_type(2))) float v2f;
typedef __attribute__((ext_vector_type(8))) float v8f;

#define B_Q      2048
#define D_DIM    1024
#define C_MEM    65536
#define TOPK     8
#define MROWS    32             // queries per block (2 WMMA M-tiles share each B frag)
#define NCHUNK   8              // grid.y of gemm kernel
#define CHUNK_COLS 8192         // C_MEM / NCHUNK
#define WAVE_COLS 1024          // CHUNK_COLS / 8 waves
#define WAVE_TILES 64           // WAVE_COLS / 16
#define LISTS_PER_Q 64          // NCHUNK * 8 waves
#define QPAD 1026               // 1024 + 2 pad (kills 16-way LDS bank conflicts, keeps b64 align)

// ---------------------------------------------------------------------------
// Kernel 1: inverse key norms  inv_kn[c] = 1 / max(||K_c||, eps)
// ---------------------------------------------------------------------------
__global__ __launch_bounds__(256) void ltm_invnorm_kernel(
    const float* __restrict__ keys, float* __restrict__ invkn) {
  const int lane = threadIdx.x & 31;
  const int wid  = threadIdx.x >> 5;
  const int c    = blockIdx.x * 8 + wid;
  const float4* row = (const float4*)(keys + (size_t)c * D_DIM);
  float ss = 0.0f;
#pragma unroll
  for (int i = 0; i < 8; ++i) {
    float4 v = row[i * 32 + lane];
    ss += v.x * v.x + v.y * v.y + v.z * v.z + v.w * v.w;
  }
#pragma unroll
  for (int off = 16; off > 0; off >>= 1) ss += __shfl_xor(ss, off, 32);
  if (lane == 0) invkn[c] = 1.0f / fmaxf(sqrtf(ss), 1e-12f);
}

// ---------------------------------------------------------------------------
// Kernel 2: fused  S = Q * K^T  (WMMA f32 16x16x4, 2 M-tiles per B fragment)
// block = 256 threads = 8 waves; block computes 32 queries x 8192 keys.
// Each wave owns a 1024-key strip (64 tiles of 16 cols) x all 32 query rows:
// per K-step it loads ONE global B fragment and TWO LDS A fragments, issuing
// two v_wmma (rows 0-15 and rows 16-31) -> 128B global traffic per WMMA.
// Lane l keeps the running top-8 for query row l in registers.
// ---------------------------------------------------------------------------
__global__ __launch_bounds__(256) void ltm_gemm_topk_kernel(
    const float* __restrict__ q, const float* __restrict__ keys,
    const float* __restrict__ invkn,
    float* __restrict__ pvals, int* __restrict__ pidx) {
  __shared__ float Qs[MROWS * QPAD];       // 32 x 1024 query tile (padded) ~128KB
  __shared__ float Ts[8 * MROWS * 17];     // per-wave 32x16 transpose staging ~17KB

  const int tid  = threadIdx.x;
  const int lane = tid & 31;
  const int wid  = tid >> 5;
  const int half = lane >> 4;           // 0: K pair {0,1}/rows 0-7 ; 1: {2,3}/rows 8-15
  const int l15  = lane & 15;

  // stage the 32x1024 query tile into padded LDS (float4 global reads)
  const float* qbase = q + (size_t)blockIdx.x * MROWS * D_DIM;
  for (int i = tid; i < MROWS * D_DIM / 4; i += 256) {
    int row = i >> 8, col4 = (i & 255) * 4;
    float4 v = ((const float4*)qbase)[i];
    float* dst = &Qs[row * QPAD + col4];
    dst[0] = v.x; dst[1] = v.y; dst[2] = v.z; dst[3] = v.w;
  }
  __syncthreads();

  // per-row top-8: lane l owns query row l of this 32-row block
  float tv[TOPK]; int ti[TOPK];
#pragma unroll
  for (int j = 0; j < TOPK; ++j) { tv[j] = -FLT_MAX; ti[j] = INT_MAX; }

  const int a0base = l15 * QPAD + 2 * half;          // A frag rows 0-15
  const int a1base = (16 + l15) * QPAD + 2 * half;   // A frag rows 16-31
  const int tsbase = wid * (MROWS * 17);

  for (int t = 0; t < WAVE_TILES; ++t) {
    const int cb = blockIdx.y * CHUNK_COLS + wid * WAVE_COLS + t * 16;
    const float* bptr = keys + (size_t)(cb + l15) * D_DIM + 2 * half;

    v8f acc0 = {};
    v8f acc1 = {};
#pragma unroll 8
    for (int k0 = 0; k0 < D_DIM; k0 += 4) {
      v2f b;  b[0]  = bptr[k0];          b[1]  = bptr[k0 + 1];
      v2f a0; a0[0] = Qs[a0base + k0];   a0[1] = Qs[a0base + k0 + 1];
      v2f a1; a1[0] = Qs[a1base + k0];   a1[1] = Qs[a1base + k0 + 1];
      acc0 = __builtin_amdgcn_wmma_f32_16x16x4_f32(
          false, a0, false, b, (short)0, acc0, false, false);
      acc1 = __builtin_amdgcn_wmma_f32_16x16x4_f32(
          false, a1, false, b, (short)0, acc1, false, false);
    }

    // scale by 1/||key|| (per column = per lane) and dump tiles through LDS
    const float s = invkn[cb + l15];
#pragma unroll
    for (int r = 0; r < 8; ++r) {
      Ts[tsbase + (half * 8 + r) * 17 + l15]        = acc0[r] * s;
      Ts[tsbase + (16 + half * 8 + r) * 17 + l15]   = acc1[r] * s;
    }
    __syncthreads();   // uniform across waves; fences DS store->load

    // every lane scans its own query row, sorted-insert into register top-8
#pragma unroll
    for (int c2 = 0; c2 < 16; ++c2) {
      float v  = Ts[tsbase + lane * 17 + c2];
      int   ci = cb + c2;
      if (v > tv[TOPK - 1]) {            // strict > keeps earlier (lower) index on ties
        tv[TOPK - 1] = v; ti[TOPK - 1] = ci;
#pragma unroll
        for (int j = TOPK - 1; j > 0; --j) {
          bool sw = (tv[j] > tv[j - 1]) ||
                    (tv[j] == tv[j - 1] && ti[j] < ti[j - 1]);
          if (sw) {
            float fv = tv[j]; tv[j] = tv[j - 1]; tv[j - 1] = fv;
            int   fi = ti[j]; ti[j] = ti[j - 1]; ti[j - 1] = fi;
          }
        }
      }
    }
  }

  // emit this wave's sorted candidate list for its 32 query rows
  {
    const int qrow = blockIdx.x * MROWS + lane;
    const int list = blockIdx.y * 8 + wid;
    const size_t base = ((size_t)qrow * LISTS_PER_Q + list) * TOPK;
#pragma unroll
    for (int j = 0; j < TOPK; ++j) { pvals[base + j] = tv[j]; pidx[base + j] = ti[j]; }
  }
}

// ---------------------------------------------------------------------------
// Kernel 3: merge 64 partial lists -> final top-8 indices per query
// ---------------------------------------------------------------------------
__global__ __launch_bounds__(256) void ltm_merge_kernel(
    const float* __restrict__ pvals, const int* __restrict__ pidx,
    int* __restrict__ fidx) {
  const int lane = threadIdx.x & 31;
  const int wid  = threadIdx.x >> 5;
  const int qq   = blockIdx.x * 8 + wid;

  float tv[TOPK]; int ti[TOPK];
#pragma unroll
  for (int j = 0; j < TOPK; ++j) { tv[j] = -FLT_MAX; ti[j] = INT_MAX; }

#pragma unroll
  for (int li = 0; li < 2; ++li) {
    const int list = lane * 2 + li;
    const size_t base = ((size_t)qq * LISTS_PER_Q + list) * TOPK;
#pragma unroll
    for (int j = 0; j < TOPK; ++j) {
      float v = pvals[base + j]; int idx = pidx[base + j];
      if (v > tv[TOPK - 1] || (v == tv[TOPK - 1] && idx < ti[TOPK - 1])) {
        tv[TOPK - 1] = v; ti[TOPK - 1] = idx;
#pragma unroll
        for (int jj = TOPK - 1; jj > 0; --jj) {
          bool sw = (tv[jj] > tv[jj - 1]) ||
                    (tv[jj] == tv[jj - 1] && ti[jj] < ti[jj - 1]);
          if (sw) {
            float fv = tv[jj]; tv[jj] = tv[jj - 1]; tv[jj - 1] = fv;
            int   fi = ti[jj]; ti[jj] = ti[jj - 1]; ti[jj - 1] = fi;
          }
        }
      }
    }
  }

#pragma unroll
  for (int r = 0; r < TOPK; ++r) {
    float v = tv[0]; int i = ti[0];
#pragma unroll
    for (int off = 16; off > 0; off >>= 1) {
      float ov = __shfl_xor(v, off, 32);
      int   oi = __shfl_xor(i, off, 32);
      if (ov > v || (ov == v && oi < i)) { v = ov; i = oi; }
    }
    if (lane == 0) fidx[qq * TOPK + r] = i;
    if (ti[0] == i) {                     // winner pops its head (indices unique)
#pragma unroll
      for (int j = 0; j < TOPK - 1; ++j) { tv[j] = tv[j + 1]; ti[j] = ti[j + 1]; }
      tv[TOPK - 1] = -FLT_MAX; ti[TOPK - 1] = INT_MAX;
    }
  }
}

// ---------------------------------------------------------------------------
// Kernel 4: gather retrieved keys/values (float4 row copies)
// ---------------------------------------------------------------------------
__global__ __launch_bounds__(256) void ltm_gather_kernel(
    const float* __restrict__ keys, const float* __restrict__ vals,
    const int* __restrict__ fidx,
    float* __restrict__ okeys, float* __restrict__ ovals) {
  const int r = blockIdx.x;              // (query, j) flattened: 0..16383
  const int idx = fidx[r];
  const int t = threadIdx.x;             // 256 threads x float4 = 1024 floats
  const float4* k4 = (const float4*)keys + (size_t)idx * 256;
  const float4* v4 = (const float4*)vals + (size_t)idx * 256;
  float4* ok4 = (float4*)okeys + (size_t)r * 256;
  float4* ov4 = (float4*)ovals + (size_t)r * 256;
  ok4[t] = k4[t];
  ov4[t] = v4[t];
}

// ---------------------------------------------------------------------------
// Kernel 5/6: usage = usage_in + counts   (copy, then exact float atomicAdds)
// ---------------------------------------------------------------------------
__global__ __launch_bounds__(256) void ltm_usage_copy_kernel(
    const float* __restrict__ uin, float* __restrict__ uout) {
  const int i = blockIdx.x * 256 + threadIdx.x;
  uout[i] = uin[i];
}

__global__ __launch_bounds__(256) void ltm_usage_scatter_kernel(
    const int* __restrict__ fidx, float* __restrict__ uout) {
  const int i = blockIdx.x * 256 + threadIdx.x;   // 0..16383
  atomicAdd(&uout[fidx[i]], 1.0f);
}

// ---------------------------------------------------------------------------
extern "C" void kernel_launch(void* const* d_in, const int* in_sizes, int n_in,
                              void* d_out, int out_size, void* d_ws, size_t ws_size,
                              hipStream_t stream) {
  (void)in_sizes; (void)n_in; (void)out_size; (void)ws_size;
  const float* query  = (const float*)d_in[0];   // [2048,1024]
  const float* keys   = (const float*)d_in[1];   // [65536,1024]
  const float* values = (const float*)d_in[2];   // [65536,1024]
  const float* usage  = (const float*)d_in[3];   // [65536]
  // d_in[4] is k; fixed at 8 by the reference setup.

  float* out       = (float*)d_out;
  float* out_keys  = out;                                   // [2048*8*1024]
  float* out_vals  = out + (size_t)B_Q * TOPK * D_DIM;      // [2048*8*1024]
  float* out_usage = out + (size_t)2 * B_Q * TOPK * D_DIM;  // [65536]

  float* invkn = (float*)d_ws;                                   // 65536 f32
  float* pvals = invkn + C_MEM;                                  // 2048*64*8 f32
  int*   pidx  = (int*)(pvals + (size_t)B_Q * LISTS_PER_Q * TOPK);
  int*   fidx  = pidx + (size_t)B_Q * LISTS_PER_Q * TOPK;        // 2048*8 int

  ltm_invnorm_kernel<<<C_MEM / 8, 256, 0, stream>>>(keys, invkn);
  ltm_gemm_topk_kernel<<<dim3(B_Q / MROWS, NCHUNK), 256, 0, stream>>>(
      query, keys, invkn, pvals, pidx);
  ltm_merge_kernel<<<B_Q / 8, 256, 0, stream>>>(pvals, pidx, fidx);
  ltm_gather_kernel<<<B_Q * TOPK, 256, 0, stream>>>(
      keys, values, fidx, out_keys, out_vals);
  ltm_usage_copy_kernel<<<C_MEM / 256, 256, 0, stream>>>(usage, out_usage);
  ltm_usage_scatter_kernel<<<B_Q * TOPK / 256, 256, 0, stream>>>(fidx, out_usage);
}